// CVRPDecoder_40398462386332
// MI455X (gfx1250) — compile-verified
//
#include <hip/hip_runtime.h>
#include <math.h>

// Output layout (float32, concatenated in reference return order):
//   [ order : N-1 ][ route_ids : N-1 ][ edge_probs : N*N ][ adj : N*N ]

typedef __attribute__((ext_vector_type(4))) float v4f;
typedef __attribute__((ext_vector_type(4))) int   v4i;

// ---------------------------------------------------------------------------
// Phase 0: zero the adjacency region (scatter semantics: .at[].set on zeros)
// ---------------------------------------------------------------------------
__global__ void cvrp_zero_adj(float* __restrict__ adj, long long nn) {
  long long i = (long long)blockIdx.x * blockDim.x + threadIdx.x;
  long long stride = (long long)gridDim.x * blockDim.x;
  long long nn4 = nn >> 2;
  v4f z = {0.0f, 0.0f, 0.0f, 0.0f};
  for (long long k = i; k < nn4; k += stride)
    reinterpret_cast<v4f*>(adj)[k] = z;
  for (long long k = (nn4 << 2) + i; k < nn; k += stride)
    adj[k] = 0.0f;
}

// ---------------------------------------------------------------------------
// Phase 1: sigmoid + scatter.  probs written non-temporal (never re-read);
// adj written regular-temporal so the 64 MB matrix stays hot in the 192 MB L2
// for the latency-bound decode phase.
// ---------------------------------------------------------------------------
__global__ void cvrp_sigmoid_scatter(const float* __restrict__ preds,
                                     const int*   __restrict__ srcv,
                                     const int*   __restrict__ dstv,
                                     float*       __restrict__ probs,
                                     float*       __restrict__ adj,
                                     long long nn, int N) {
  long long i = (long long)blockIdx.x * blockDim.x + threadIdx.x;
  long long stride = (long long)gridDim.x * blockDim.x;
  long long nn4 = nn >> 2;
  for (long long k = i; k < nn4; k += stride) {
    v4f p = reinterpret_cast<const v4f*>(preds)[k];
    v4i s = reinterpret_cast<const v4i*>(srcv)[k];
    v4i d = reinterpret_cast<const v4i*>(dstv)[k];
    v4f q;
    q.x = 1.0f / (1.0f + __expf(-p.x));
    q.y = 1.0f / (1.0f + __expf(-p.y));
    q.z = 1.0f / (1.0f + __expf(-p.z));
    q.w = 1.0f / (1.0f + __expf(-p.w));
    __builtin_nontemporal_store(q, reinterpret_cast<v4f*>(probs) + k);
    adj[(long long)s.x * N + d.x] = q.x;   // RT stores: keep adj in L2
    adj[(long long)s.y * N + d.y] = q.y;
    adj[(long long)s.z * N + d.z] = q.z;
    adj[(long long)s.w * N + d.w] = q.w;
  }
  for (long long k = (nn4 << 2) + i; k < nn; k += stride) {
    float q = 1.0f / (1.0f + __expf(-preds[k]));
    probs[k] = q;
    adj[(long long)srcv[k] * N + dstv[k]] = q;
  }
}

// ---------------------------------------------------------------------------
// Phase 2: sequential greedy decode.  One workgroup, 32 wave32 waves.
// Per step: async-DMA the current row (16 KB) into LDS with
// GLOBAL_LOAD_ASYNC_TO_LDS_B128 (one instruction per wave = 512 B, ASYNCcnt),
// masked argmax as a monotonic u64 key (value bits << 32 | ~index), wave32
// shuffle tree, then one ds_max_u64 per wave leader into a single LDS slot.
// Key packing makes max(key) == (largest prob, lowest index on ties), which
// matches jnp.argmax exactly (probs are non-negative so IEEE bits are
// order-preserving).
// ---------------------------------------------------------------------------
#define DEC_THREADS 1024

__global__ __launch_bounds__(DEC_THREADS)
void cvrp_greedy_decode(const float* __restrict__ adj,
                        float*       __restrict__ order_out,
                        float*       __restrict__ route_out,
                        const int*   __restrict__ cap_ptr,
                        int N) {
  __shared__ v4f                rowbuf[DEC_THREADS];     // 16 KB row tile
  __shared__ unsigned char      visited[4 * DEC_THREADS];
  __shared__ unsigned long long s_best;                  // packed (val,~idx)
  __shared__ int                s_state[3];              // current, rlen, rid

  const int tid  = threadIdx.x;
  const int lane = tid & 31;
  const int cap  = *cap_ptr;

  for (int i = tid; i < N; i += DEC_THREADS) visited[i] = 0;
  if (tid == 0) {
    visited[0] = 1;
    s_state[0] = 0; s_state[1] = 0; s_state[2] = 0;
    s_best = 0ull;
  }
  __syncthreads();

  // LDS byte offset of this lane's 16B slot (flat LDS addr: low 32 bits = offset)
  const unsigned lds_off = (unsigned)(size_t)&rowbuf[tid];

  const int steps = N - 1;
  for (int step = 0; step < steps; ++step) {
    const int current = s_state[0];

    // ---- async DMA: 4 floats per lane, whole row staged in LDS ----
    unsigned long long ga =
        (unsigned long long)(adj + (size_t)current * (size_t)N + (size_t)tid * 4);
    asm volatile("global_load_async_to_lds_b128 %0, %1, off"
                 :: "v"(lds_off), "v"(ga) : "memory");
    asm volatile("s_wait_asynccnt 0x0" ::: "memory");

    // ---- masked per-thread argmax over own 4 elements as packed u64 key ----
    v4f v = rowbuf[tid];                                   // own-wave data only
    unsigned m = *reinterpret_cast<const unsigned*>(&visited[tid * 4]);
    float x0 = ((m      ) & 0xffu) ? 0.0f : v.x;
    float x1 = ((m >>  8) & 0xffu) ? 0.0f : v.y;
    float x2 = ((m >> 16) & 0xffu) ? 0.0f : v.z;
    float x3 = ((m >> 24) & 0xffu) ? 0.0f : v.w;
    const unsigned base = (unsigned)(tid * 4);
    unsigned long long k0 =
        ((unsigned long long)__float_as_uint(x0) << 32) | (0xffffffffu - base);
    unsigned long long k1 =
        ((unsigned long long)__float_as_uint(x1) << 32) | (0xffffffffu - (base + 1));
    unsigned long long k2 =
        ((unsigned long long)__float_as_uint(x2) << 32) | (0xffffffffu - (base + 2));
    unsigned long long k3 =
        ((unsigned long long)__float_as_uint(x3) << 32) | (0xffffffffu - (base + 3));
    unsigned long long key = k0 > k1 ? k0 : k1;
    unsigned long long k23 = k2 > k3 ? k2 : k3;
    key = key > k23 ? key : k23;

    // ---- wave32 shuffle reduction on the packed key ----
    #pragma unroll
    for (int off = 16; off > 0; off >>= 1) {
      unsigned long long other =
          (unsigned long long)__shfl_xor((long long)key, off, 32);
      key = other > key ? other : key;
    }
    // ---- one ds_max_u64 per wave leader into a single LDS slot ----
    if (lane == 0) atomicMax(&s_best, key);
    __syncthreads();

    // ---- scalar state update (thread 0) ----
    if (tid == 0) {
      unsigned long long best = s_best;
      float bv = __uint_as_float((unsigned)(best >> 32));
      int   bi = (int)(0xffffffffu - (unsigned)best);
      const bool valid = (bv > 0.0f) && (bi < N);
      int rlen = s_state[1];
      int rid  = s_state[2];
      if (valid) { visited[bi] = 1; rlen += 1; }
      const bool end_route = (!valid) || (rlen >= cap);
      order_out[step] = valid ? (float)bi : -1.0f;
      route_out[step] = (float)rid;                 // rid BEFORE increment
      s_state[0] = end_route ? 0 : bi;              // restart at depot
      s_state[1] = end_route ? 0 : rlen;
      s_state[2] = rid + (end_route ? 1 : 0);
      s_best = 0ull;                                // re-arm for next step
    }
    __syncthreads();
  }
}

// ---------------------------------------------------------------------------
extern "C" void kernel_launch(void* const* d_in, const int* in_sizes, int n_in,
                              void* d_out, int out_size, void* d_ws, size_t ws_size,
                              hipStream_t stream) {
  const long long nn = (long long)in_sizes[0];              // N*N
  const int N = (int)llround(sqrt((double)nn));

  const float* preds = (const float*)d_in[0];
  const int*   ei    = (const int*)d_in[1];                 // [2, N*N]
  const int*   srcv  = ei;
  const int*   dstv  = ei + nn;
  const int*   cap   = (const int*)d_in[3];                 // device scalar

  float* out   = (float*)d_out;
  float* order = out;
  float* route = out + (N - 1);
  float* probs = out + 2 * (size_t)(N - 1);
  float* adj   = probs + nn;

  cvrp_zero_adj<<<2048, 256, 0, stream>>>(adj, nn);
  cvrp_sigmoid_scatter<<<4096, 256, 0, stream>>>(preds, srcv, dstv, probs, adj, nn, N);
  cvrp_greedy_decode<<<1, DEC_THREADS, 0, stream>>>(adj, order, route, cap, N);
}